// Multi_Task_6184752906505
// MI455X (gfx1250) — compile-verified
//
#include <hip/hip_runtime.h>
#include <math.h>

// Problem constants (from reference): B=4096, T=64, D=256, H=128, K=12
#define B_DIM 4096
#define T_DIM 64
#define D_DIM 256
#define H_DIM 128
#define K_DIM 12

typedef float v2f __attribute__((ext_vector_type(2)));
typedef float v8f __attribute__((ext_vector_type(8)));

// Full-precision f32 WMMA: D(16x16) = A(16x4) * B(4x16) + C
__device__ __forceinline__ v8f wmma_f32(v2f a, v2f b, v8f c) {
    return __builtin_amdgcn_wmma_f32_16x16x4_f32(
        /*neg_a=*/false, a, /*neg_b=*/false, b,
        /*c_mod=*/(short)0, c, /*reuse_a=*/false, /*reuse_b=*/false);
}

// CDNA5 async global->LDS copy (ASYNCcnt tracked; no VGPR data staging).
// Low 32 bits of a generic pointer to __shared__ = wave-relative LDS offset.
__device__ __forceinline__ void async_load_b128(const float* gsrc, float* ldst) {
    unsigned loff = (unsigned)(unsigned long long)ldst;
    asm volatile("global_load_async_to_lds_b128 %0, %1, off"
                 :: "v"(loff), "v"(gsrc) : "memory");
}
__device__ __forceinline__ void wait_async0() {
    asm volatile("s_wait_asynccnt 0x0" ::: "memory");
}

// ---------------------------------------------------------------------------
// Stage 1: per-molecule atom gating + pooling.
//   logits(16x64) = w_atom_pad(16x256) * f_b^T(256x64)
//   aw = sigmoid(logits + b_atom) * valid      -> atom_weights output
//   mol(16x256)  = aw(16x64) * f_b(64x256)     -> workspace
// One 256-thread block (8 waves) per molecule b.
// ---------------------------------------------------------------------------
#define SF_STRIDE 260
#define SAW_STRIDE 68

__global__ __launch_bounds__(256) void atoms_kernel(
    const int* __restrict__ data_x, const float* __restrict__ feats,
    const float* __restrict__ w_atom, const float* __restrict__ b_atom,
    float* __restrict__ atomw_out, float* __restrict__ mol)
{
    __shared__ float sf[T_DIM][SF_STRIDE];      // f[t][d]
    __shared__ float sw[16][SF_STRIDE];         // w_atom, rows 12..15 zero
    __shared__ float sLog[2][16][SAW_STRIDE];   // depth-half partials
    __shared__ float sAW[16][SAW_STRIDE];       // masked sigmoid weights
    __shared__ float sValid[T_DIM];
    __shared__ float sbat[16];

    const int tid = threadIdx.x;
    const int b = blockIdx.x;
    const int lane = tid & 31;
    const int wv = tid >> 5;
    const int laneLo = lane & 15;
    const int laneHalf = lane >> 4;

    // --- async copy f_b = feats[b, 1:, :] (64x256 f32) into LDS ---
    {
        const float* fb = feats + ((size_t)b * (T_DIM + 1) + 1) * D_DIM;
        const int fr = tid >> 6;            // 4 rows per pass, 64 f4 per row
        const int fc = (tid & 63) * 4;
        #pragma unroll
        for (int i = 0; i < 16; ++i) {
            int row = fr + 4 * i;
            async_load_b128(fb + (size_t)row * D_DIM + fc, &sf[row][fc]);
        }
    }
    // --- w_atom padded to 16 rows (regular load; needs zero fill) ---
    {
        #pragma unroll
        for (int i = 0; i < 4; ++i) {
            int fi = tid + 256 * i;        // 0..1023, 64 float4 per row
            int row = fi >> 6;
            int c = (fi & 63) * 4;
            float4 v = make_float4(0.f, 0.f, 0.f, 0.f);
            if (row < K_DIM)
                v = *(const float4*)(w_atom + (size_t)row * D_DIM + c);
            *(float4*)(&sw[row][c]) = v;
        }
        if (tid < 16) sbat[tid] = (tid < K_DIM) ? b_atom[tid] : 0.f;
        if (tid < T_DIM)
            sValid[tid] = (data_x[(size_t)b * T_DIM + tid] != 0) ? 1.f : 0.f;
    }
    wait_async0();
    __syncthreads();

    // --- logits via WMMA: wave w -> t-tile (w&3), depth half (w>>2) ---
    {
        const int t0 = (wv & 3) * 16;
        const int d0 = (wv >> 2) * 128;
        v8f acc = {};
        for (int d = d0; d < d0 + 128; d += 4) {
            const int dr = d + 2 * laneHalf;
            v2f a, bf;
            a.x = sw[laneLo][dr];          a.y = sw[laneLo][dr + 1];
            bf.x = sf[t0 + laneLo][dr];    bf.y = sf[t0 + laneLo][dr + 1];
            acc = wmma_f32(a, bf, acc);
        }
        #pragma unroll
        for (int j = 0; j < 8; ++j)
            sLog[wv >> 2][j + 8 * laneHalf][t0 + laneLo] = acc[j];
    }
    __syncthreads();

    // --- combine halves, sigmoid, mask; emit atom_weights ---
    for (int idx = tid; idx < 16 * T_DIM; idx += 256) {
        int kk = idx >> 6;
        int t = idx & 63;
        float l = sLog[0][kk][t] + sLog[1][kk][t] + sbat[kk];
        float s = 1.f / (1.f + __expf(-l));
        float aw = s * sValid[t];
        sAW[kk][t] = aw;
        if (kk < K_DIM)
            atomw_out[((size_t)kk * B_DIM + b) * T_DIM + t] = aw;
    }
    __syncthreads();

    // --- mol via WMMA: wave w -> d-tiles {32w, 32w+16} ---
    {
        const int n0 = wv * 32;
        v8f acc0 = {}, acc1 = {};
        for (int t = 0; t < T_DIM; t += 4) {
            const int tr = t + 2 * laneHalf;
            v2f a, b0, b1;
            a.x = sAW[laneLo][tr];              a.y = sAW[laneLo][tr + 1];
            b0.x = sf[tr][n0 + laneLo];         b0.y = sf[tr + 1][n0 + laneLo];
            b1.x = sf[tr][n0 + 16 + laneLo];    b1.y = sf[tr + 1][n0 + 16 + laneLo];
            acc0 = wmma_f32(a, b0, acc0);
            acc1 = wmma_f32(a, b1, acc1);
        }
        #pragma unroll
        for (int j = 0; j < 8; ++j) {
            int kk = j + 8 * laneHalf;
            if (kk < K_DIM) {
                float* mo = mol + ((size_t)kk * B_DIM + b) * D_DIM;
                mo[n0 + laneLo] = acc0[j];
                mo[n0 + 16 + laneLo] = acc1[j];
            }
        }
    }
}

// ---------------------------------------------------------------------------
// Pure batched GEMM + bias + ReLU + column stats, async double-buffered LDS.
//   out[k] = relu( A[k] @ Bm[k] + bias[k] )
// (Previous layer's BN affine is pre-folded into Bm/bias by wprep_kernel.)
// Block: 256 threads = 8 waves; tile 64(M) x 128(N); depth chunked by 64.
// ---------------------------------------------------------------------------
__global__ __launch_bounds__(256) void gemm_relu_stats_kernel(
    const float* __restrict__ A, const float* __restrict__ Bm,
    const float* __restrict__ bias, float* __restrict__ out,
    float* __restrict__ stSum, float* __restrict__ stSq, int Kd)
{
    const int N = H_DIM;
    __shared__ float sA[2][64][68];
    __shared__ float sB[2][64][132];
    __shared__ float sSum[H_DIM];
    __shared__ float sSq[H_DIM];

    const int tid = threadIdx.x;
    const int lane = tid & 31;
    const int wv = tid >> 5;
    const int laneLo = lane & 15;
    const int laneHalf = lane >> 4;
    const int k = blockIdx.y;
    const int rowBase = blockIdx.x * 64;

    if (tid < H_DIM) { sSum[tid] = 0.f; sSq[tid] = 0.f; }

    const int rT = (wv >> 1) * 16;   // row tile within block (0,16,32,48)
    const int cG = (wv & 1) * 64;    // column group (0 or 64)

    v8f acc[4] = {{}, {}, {}, {}};

    const float* Ak = A + ((size_t)k * B_DIM + rowBase) * Kd;
    const float* Bk = Bm + (size_t)k * Kd * N;

    // per-thread async slots
    const int ar = tid >> 4, ac = (tid & 15) * 4;   // A: 16 f4/row
    const int br = tid >> 5, bc = (tid & 31) * 4;   // B: 32 f4/row

    auto issue_chunk = [&](int kc, int buf) {
        #pragma unroll
        for (int i = 0; i < 4; ++i) {
            int row = ar + 16 * i;
            async_load_b128(Ak + (size_t)row * Kd + kc + ac, &sA[buf][row][ac]);
        }
        #pragma unroll
        for (int i = 0; i < 8; ++i) {
            int row = br + 8 * i;
            async_load_b128(Bk + (size_t)(kc + row) * N + bc, &sB[buf][row][bc]);
        }
    };

    const int nChunks = Kd >> 6;
    issue_chunk(0, 0);
    wait_async0();
    __syncthreads();

    for (int c = 0; c < nChunks; ++c) {
        const int cur = c & 1;
        if (c + 1 < nChunks)
            issue_chunk((c + 1) << 6, cur ^ 1);   // prefetch next chunk async

        #pragma unroll 4
        for (int kk = 0; kk < 64; kk += 4) {
            const int kr = kk + 2 * laneHalf;
            v2f a;
            a.x = sA[cur][rT + laneLo][kr];
            a.y = sA[cur][rT + laneLo][kr + 1];
            #pragma unroll
            for (int ct = 0; ct < 4; ++ct) {
                v2f bf;
                bf.x = sB[cur][kr][cG + ct * 16 + laneLo];
                bf.y = sB[cur][kr + 1][cG + ct * 16 + laneLo];
                acc[ct] = wmma_f32(a, bf, acc[ct]);
            }
        }
        wait_async0();
        __syncthreads();
    }

    // Epilogue: bias + ReLU + store + LDS stats reduction
    #pragma unroll
    for (int ct = 0; ct < 4; ++ct) {
        const int col = cG + ct * 16 + laneLo;
        const float bia = bias[(size_t)k * N + col];
        float ls = 0.f, ls2 = 0.f;
        #pragma unroll
        for (int j = 0; j < 8; ++j) {
            int row = rowBase + rT + j + 8 * laneHalf;
            float v = fmaxf(acc[ct][j] + bia, 0.f);
            out[((size_t)k * B_DIM + row) * N + col] = v;
            ls += v;
            ls2 += v * v;
        }
        atomicAdd(&sSum[col], ls);    // ds_add_f32
        atomicAdd(&sSq[col], ls2);
    }
    __syncthreads();
    if (tid < H_DIM) {
        atomicAdd(&stSum[(size_t)k * N + tid], sSum[tid]);   // global_atomic_add_f32
        atomicAdd(&stSq[(size_t)k * N + tid], sSq[tid]);
    }
}

// ---------------------------------------------------------------------------
// BN finalize: (sum, sumsq) -> (alpha = rstd*gamma, beta' = be - mean*alpha)
// ---------------------------------------------------------------------------
__global__ void bn_finalize_kernel(
    const float* __restrict__ sum, const float* __restrict__ sq,
    const float* __restrict__ g, const float* __restrict__ be,
    float* __restrict__ alpha, float* __restrict__ beta)
{
    int idx = blockIdx.x * blockDim.x + threadIdx.x;
    if (idx >= K_DIM * H_DIM) return;
    const float invB = 1.f / (float)B_DIM;
    float m = sum[idx] * invB;
    float var = sq[idx] * invB - m * m;
    float a = rsqrtf(var + 1e-5f) * g[idx];
    alpha[idx] = a;
    beta[idx] = be[idx] - m * a;
}

// ---------------------------------------------------------------------------
// Fold BN affine of the *input* into the next layer's weights/bias:
//   (x*alpha + beta) @ W + b  ==  x @ (diag(alpha)W) + (beta^T W + b)
// One block per k, 128 threads = one output column each.
// ---------------------------------------------------------------------------
__global__ __launch_bounds__(128) void wprep_kernel(
    const float* __restrict__ W, const float* __restrict__ bias,
    const float* __restrict__ alpha, const float* __restrict__ beta,
    float* __restrict__ Weff, float* __restrict__ biasEff)
{
    const int k = blockIdx.x;
    const int n = threadIdx.x;
    const float* Wk = W + (size_t)k * H_DIM * H_DIM;
    const float* al = alpha + (size_t)k * H_DIM;
    const float* be = beta + (size_t)k * H_DIM;
    float* We = Weff + (size_t)k * H_DIM * H_DIM;
    float acc = bias[(size_t)k * H_DIM + n];
    for (int d = 0; d < H_DIM; ++d) {
        float w = Wk[(size_t)d * H_DIM + n];
        We[(size_t)d * H_DIM + n] = al[d] * w;
        acc += be[d] * w;
    }
    biasEff[(size_t)k * H_DIM + n] = acc;
}

__global__ void zero_kernel(float* __restrict__ p, int n) {
    int idx = blockIdx.x * blockDim.x + threadIdx.x;
    if (idx < n) p[idx] = 0.f;
}

// ---------------------------------------------------------------------------
// Final prediction: one wave per (k,b). Fold BN3 affine into the Wout dot.
// ---------------------------------------------------------------------------
__global__ __launch_bounds__(256) void pred_kernel(
    const float* __restrict__ h3, const float* __restrict__ alpha,
    const float* __restrict__ beta, const float* __restrict__ Wout,
    const float* __restrict__ bout, float* __restrict__ outPred)
{
    const int wv = threadIdx.x >> 5;
    const int lane = threadIdx.x & 31;
    const size_t g = (size_t)blockIdx.x * 8 + wv;   // 0 .. K*B-1
    const int k = (int)(g / B_DIM);
    const int b = (int)(g % B_DIM);

    const float* row = h3 + ((size_t)k * B_DIM + b) * H_DIM;
    const float* al = alpha + (size_t)k * H_DIM;
    const float* be = beta + (size_t)k * H_DIM;
    const float* w = Wout + (size_t)k * H_DIM;

    float p = 0.f;
    #pragma unroll
    for (int i = 0; i < 4; ++i) {
        int h = lane + 32 * i;
        p += (row[h] * al[h] + be[h]) * w[h];
    }
    #pragma unroll
    for (int ofs = 16; ofs > 0; ofs >>= 1)
        p += __shfl_xor(p, ofs, 32);
    if (lane == 0)
        outPred[(size_t)b * K_DIM + k] = p + bout[k];
}

// ---------------------------------------------------------------------------
extern "C" void kernel_launch(void* const* d_in, const int* in_sizes, int n_in,
                              void* d_out, int out_size, void* d_ws, size_t ws_size,
                              hipStream_t stream) {
    (void)in_sizes; (void)n_in; (void)out_size; (void)ws_size;

    const int* data_x = (const int*)d_in[0];
    const float* feats = (const float*)d_in[1];
    const float* w_atom = (const float*)d_in[2];
    const float* b_atom = (const float*)d_in[3];
    const float* W1 = (const float*)d_in[4];
    const float* b1 = (const float*)d_in[5];
    const float* g1 = (const float*)d_in[6];
    const float* be1 = (const float*)d_in[7];
    const float* W2 = (const float*)d_in[8];
    const float* b2 = (const float*)d_in[9];
    const float* g2 = (const float*)d_in[10];
    const float* be2 = (const float*)d_in[11];
    const float* W3 = (const float*)d_in[12];
    const float* b3 = (const float*)d_in[13];
    const float* g3 = (const float*)d_in[14];
    const float* be3 = (const float*)d_in[15];
    const float* Wout = (const float*)d_in[16];
    const float* bout = (const float*)d_in[17];

    float* out = (float*)d_out;
    float* pred = out;                               // (B,K) = 49152 floats
    float* atomw = out + (size_t)B_DIM * K_DIM;      // (K,B,T)

    // Workspace layout (floats)
    float* ws = (float*)d_ws;
    const size_t molN = (size_t)K_DIM * B_DIM * D_DIM;   // 12.58M
    const size_t hN = (size_t)K_DIM * B_DIM * H_DIM;     // 6.29M
    const size_t statN = (size_t)K_DIM * H_DIM;          // 1536
    float* mol = ws;
    float* h1 = ws + molN;
    float* h2 = h1 + hN;
    float* h3 = mol;                    // mol is dead after stage 2: alias
    float* stats = h2 + hN;             // 6 arrays
    float* sum1 = stats + 0 * statN;
    float* sq1  = stats + 1 * statN;
    float* sum2 = stats + 2 * statN;
    float* sq2  = stats + 3 * statN;
    float* sum3 = stats + 4 * statN;
    float* sq3  = stats + 5 * statN;
    float* ab = stats + 6 * statN;      // alpha/beta pairs
    float* a1 = ab + 0 * statN; float* bn1 = ab + 1 * statN;
    float* a2 = ab + 2 * statN; float* bn2 = ab + 3 * statN;
    float* a3 = ab + 4 * statN; float* bn3 = ab + 5 * statN;
    float* Weff = ab + 6 * statN;                       // K*H*H (reused for L2,L3)
    float* biasEff = Weff + (size_t)K_DIM * H_DIM * H_DIM;

    // 0) zero BN stat accumulators (atomics -> must reset every launch)
    {
        int n = (int)(6 * statN);
        zero_kernel<<<(n + 255) / 256, 256, 0, stream>>>(stats, n);
    }
    // 1) atom gating + pooling
    atoms_kernel<<<B_DIM, 256, 0, stream>>>(data_x, feats, w_atom, b_atom, atomw, mol);
    // 2) layer 1: (4096x256)x(256x128)
    gemm_relu_stats_kernel<<<dim3(B_DIM / 64, K_DIM), 256, 0, stream>>>(
        mol, W1, b1, h1, sum1, sq1, D_DIM);
    bn_finalize_kernel<<<6, 256, 0, stream>>>(sum1, sq1, g1, be1, a1, bn1);
    // 3) layer 2: BN1 folded into W2/b2, then pure GEMM
    wprep_kernel<<<K_DIM, 128, 0, stream>>>(W2, b2, a1, bn1, Weff, biasEff);
    gemm_relu_stats_kernel<<<dim3(B_DIM / 64, K_DIM), 256, 0, stream>>>(
        h1, Weff, biasEff, h2, sum2, sq2, H_DIM);
    bn_finalize_kernel<<<6, 256, 0, stream>>>(sum2, sq2, g2, be2, a2, bn2);
    // 4) layer 3: BN2 folded into W3/b3
    wprep_kernel<<<K_DIM, 128, 0, stream>>>(W3, b3, a2, bn2, Weff, biasEff);
    gemm_relu_stats_kernel<<<dim3(B_DIM / 64, K_DIM), 256, 0, stream>>>(
        h2, Weff, biasEff, h3, sum3, sq3, H_DIM);
    bn_finalize_kernel<<<6, 256, 0, stream>>>(sum3, sq3, g3, be3, a3, bn3);
    // 5) prediction head: BN3 folded into Wout dot
    pred_kernel<<<(K_DIM * B_DIM) / 8, 256, 0, stream>>>(h3, a3, bn3, Wout, bout, pred);
}